// RNNLayer_57380763074660
// MI455X (gfx1250) — compile-verified
//
#include <hip/hip_runtime.h>

// LSTM-with-resets for MI455X (gfx1250), fp32 via V_WMMA_F32_16X16X4_F32,
// with the shared B panel staged into LDS via the gfx1250 async global->LDS
// datapath (ASYNCcnt) and double-buffered against the WMMA inner loop.
//
// Phase 1: xg[T*B, 4H] = X[T*B, D] @ W[0:D, 4H] + bias   (parallel WMMA GEMM)
// Phase 2: per t: gates = xg[t] + (h_{t-1} * (1-r_t)) @ W[D:D+H, 4H],
//          fused elementwise LSTM cell update; h_t written straight into d_out.
// Phase 3: hT = out[T-1], cT = c_state (async d2d copies).

typedef __attribute__((ext_vector_type(2))) float v2f;
typedef __attribute__((ext_vector_type(4))) float v4f;
typedef __attribute__((ext_vector_type(8))) float v8f;
typedef __attribute__((ext_vector_type(4))) int   v4i;

typedef __attribute__((address_space(1))) v4i* glob_b128_t;  // AS1 int4*
typedef __attribute__((address_space(3))) v4i* lds_b128_t;   // AS3 int4*

#define T_DIM 128
#define B_DIM 128
#define D_DIM 1024
#define H_DIM 1024
#define N4    (4 * H_DIM)      // 4096 gate columns
#define KC    64               // K-chunk staged in LDS
#define NC    (1024 / KC)      // 16 chunks (D and H are both 1024)

#if defined(__gfx1250__) && __has_builtin(__builtin_amdgcn_global_load_async_to_lds_b128)
#define ASYNC_LDS 1
#else
#define ASYNC_LDS 0
#endif

__device__ __forceinline__ void async_wait0() {
#if ASYNC_LDS
#if __has_builtin(__builtin_amdgcn_s_wait_asynccnt)
    __builtin_amdgcn_s_wait_asynccnt(0);
#else
    asm volatile("s_wait_asynccnt 0" ::: "memory");
#endif
#endif
}

// Copy 4 consecutive floats global -> LDS (async DMA when available).
__device__ __forceinline__ void stage4(const float* __restrict__ gsrc,
                                       float* __restrict__ ldst) {
#if ASYNC_LDS
    __builtin_amdgcn_global_load_async_to_lds_b128(
        (glob_b128_t)gsrc, (lds_b128_t)ldst, 0, 0);
#else
    *(v4f*)ldst = *(const v4f*)gsrc;   // global_load_b128 + ds_store_b128
#endif
}

__device__ __forceinline__ float sigmoid_f(float x) {
    return 1.0f / (1.0f + __expf(-x));
}
__device__ __forceinline__ float tanh_f(float x) {
    float e = __expf(-2.0f * x);
    return (1.0f - e) / (1.0f + e);
}

// Stage a KC x 64 tile of W rows [k0, k0+KC) x cols [n0, n0+64) into buf[kr*64+c].
__device__ __forceinline__ void issue_chunk_cols(const float* __restrict__ W,
                                                 int k0, int n0,
                                                 float* __restrict__ buf, int tid) {
#pragma unroll
    for (int p = 0; p < 4; ++p) {
        const int e  = p * 1024 + tid * 4;   // 4096 floats per chunk
        const int kr = e >> 6;
        const int c  = e & 63;
        stage4(W + (size_t)(k0 + kr) * N4 + n0 + c, buf + kr * 64 + c);
    }
}

// Stage KC rows of the 4 gate strips (16 cols each) for hidden tile n0:
// buf[kr*64 + g*16 + j] = W[(D + k0 + kr)*N4 + g*H + n0 + j]
__device__ __forceinline__ void issue_chunk_gates(const float* __restrict__ W,
                                                  int k0, int n0,
                                                  float* __restrict__ buf, int tid) {
#pragma unroll
    for (int p = 0; p < 4; ++p) {
        const int e  = p * 1024 + tid * 4;
        const int kr = e >> 6;
        const int c  = e & 63;
        const int g  = c >> 4;
        const int j  = c & 15;
        stage4(W + (size_t)(D_DIM + k0 + kr) * N4 + g * H_DIM + n0 + j,
               buf + kr * 64 + c);
    }
}

// ---------------------------------------------------------------------------
// Phase 1: xg = X @ Wx + b.  Block = 8 waves = 128 rows x 64 cols; the B panel
// (shared by all 8 waves) lives in LDS, double-buffered via async DMA.
// grid = (M/128, N4/64), block = 256 threads.
// ---------------------------------------------------------------------------
__global__ void __launch_bounds__(256) xgate_gemm(const float* __restrict__ X,
                                                  const float* __restrict__ W,
                                                  const float* __restrict__ bias,
                                                  float* __restrict__ xg) {
    __shared__ float sB[2][KC * 64];

    const int tid  = threadIdx.x;
    const int lane = tid & 31;
    const int wave = tid >> 5;
    const int l16  = lane & 15;
    const int half = lane >> 4;   // 0: K{0,1}/M 0..7 ; 1: K{2,3}/M 8..15
    const int kh   = half * 2;

    const int m0 = (blockIdx.x * 8 + wave) * 16;
    const int n0 = blockIdx.y * 64;

    const float* aptr = X + (size_t)(m0 + l16) * D_DIM + kh;

    v8f acc[4] = {};

    issue_chunk_cols(W, 0, n0, sB[0], tid);

    for (int c = 0; c < NC; ++c) {
        const int p = c & 1;
        async_wait0();
        __syncthreads();                 // chunk c resident; buf p^1 free
        if (c + 1 < NC) issue_chunk_cols(W, (c + 1) * KC, n0, sB[p ^ 1], tid);

        const float* ap = aptr + c * KC;
        const float* bk = &sB[p][kh * 64 + l16];
#pragma unroll 4
        for (int kr = 0; kr < KC; kr += 4) {
            v2f a = *(const v2f*)(ap + kr);
            const float* bkk = bk + kr * 64;
#pragma unroll
            for (int nt = 0; nt < 4; ++nt) {
                v2f b;
                b.x = bkk[nt * 16];        // ds_load_b32 (imm offsets)
                b.y = bkk[64 + nt * 16];
                acc[nt] = __builtin_amdgcn_wmma_f32_16x16x4_f32(
                    false, a, false, b, (short)0, acc[nt], false, false);
            }
        }
    }

#pragma unroll
    for (int nt = 0; nt < 4; ++nt) {
        const int col = n0 + nt * 16 + l16;
        const float bs = bias[col];
#pragma unroll
        for (int r = 0; r < 8; ++r) {
            const int rowM = m0 + half * 8 + r;   // C/D layout: VGPR r -> M
            xg[(size_t)rowM * N4 + col] = acc[nt][r] + bs;
        }
    }
}

// ---------------------------------------------------------------------------
// Phase 2: one timestep.  Block = 8 waves = all 8 batch tiles of one hidden
// tile; each wave owns the same 16x16 tile across ALL FOUR gates so the LSTM
// cell update is wave-local.  The 4 gate strips of W_h are staged in LDS.
// grid = 64 blocks x 256 threads.
// ---------------------------------------------------------------------------
__global__ void __launch_bounds__(256) lstm_step(const float* __restrict__ hprev,
                                                 const float* __restrict__ W,
                                                 const float* __restrict__ xg_t,
                                                 const float* __restrict__ reset_t,
                                                 float* __restrict__ c_state,
                                                 float* __restrict__ h_out) {
    __shared__ float sB[2][KC * 64];

    const int tid  = threadIdx.x;
    const int lane = tid & 31;
    const int wave = tid >> 5;
    const int l16  = lane & 15;
    const int half = lane >> 4;
    const int kh   = half * 2;

    const int m0 = wave * 16;           // batch tile (8 per block)
    const int n0 = blockIdx.x * 16;     // hidden tile

    const int arow = m0 + l16;
    const float amask = 1.0f - reset_t[arow];   // episode reset on h
    const float* aptr = hprev + (size_t)arow * H_DIM + kh;

    v8f acc[4] = {};

    issue_chunk_gates(W, 0, n0, sB[0], tid);

    for (int c = 0; c < NC; ++c) {
        const int p = c & 1;
        async_wait0();
        __syncthreads();
        if (c + 1 < NC) issue_chunk_gates(W, (c + 1) * KC, n0, sB[p ^ 1], tid);

        const float* ap = aptr + c * KC;
        const float* bk = &sB[p][kh * 64 + l16];
#pragma unroll 4
        for (int kr = 0; kr < KC; kr += 4) {
            v2f a = *(const v2f*)(ap + kr);
            a.x *= amask;
            a.y *= amask;
            const float* bkk = bk + kr * 64;
#pragma unroll
            for (int g = 0; g < 4; ++g) {
                v2f b;
                b.x = bkk[g * 16];
                b.y = bkk[64 + g * 16];
                acc[g] = __builtin_amdgcn_wmma_f32_16x16x4_f32(
                    false, a, false, b, (short)0, acc[g], false, false);
            }
        }
    }

    const int col = n0 + l16;
#pragma unroll
    for (int r = 0; r < 8; ++r) {
        const int rowM = m0 + half * 8 + r;
        const size_t gbase = (size_t)rowM * N4;
        const float iv = acc[0][r] + xg_t[gbase + 0 * H_DIM + col];
        const float gv = acc[1][r] + xg_t[gbase + 1 * H_DIM + col];
        const float fv = acc[2][r] + xg_t[gbase + 2 * H_DIM + col];
        const float ov = acc[3][r] + xg_t[gbase + 3 * H_DIM + col];

        const float i_s = sigmoid_f(iv);
        const float g_t = tanh_f(gv);
        const float f_s = sigmoid_f(fv + 1.0f);        // forget bias +1
        const float o_s = sigmoid_f(ov);

        const float cm = 1.0f - reset_t[rowM];         // episode reset on c
        const size_t cidx = (size_t)rowM * H_DIM + col;
        const float c_new = f_s * (c_state[cidx] * cm) + i_s * g_t;
        c_state[cidx] = c_new;
        h_out[cidx]   = o_s * tanh_f(c_new);
    }
}

// ---------------------------------------------------------------------------
extern "C" void kernel_launch(void* const* d_in, const int* in_sizes, int n_in,
                              void* d_out, int out_size, void* d_ws, size_t ws_size,
                              hipStream_t stream) {
    (void)in_sizes; (void)n_in; (void)out_size; (void)ws_size;

    const float* x     = (const float*)d_in[0];  // [T, B, D]
    const float* h0    = (const float*)d_in[1];  // [1, B, H]
    const float* c0    = (const float*)d_in[2];  // [1, B, H]
    const float* reset = (const float*)d_in[3];  // [T, B]
    const float* W     = (const float*)d_in[4];  // [D+H, 4H]
    const float* bias  = (const float*)d_in[5];  // [4H]

    float* out = (float*)d_out;                  // [T*B*H] ++ hT[B*H] ++ cT[B*H]

    float* xg      = (float*)d_ws;                             // T*B*4H floats
    float* c_state = xg + (size_t)T_DIM * B_DIM * N4;          // B*H floats

    (void)hipMemcpyAsync(c_state, c0, sizeof(float) * B_DIM * H_DIM,
                         hipMemcpyDeviceToDevice, stream);

    dim3 blk(256, 1, 1);

    dim3 g1(T_DIM * B_DIM / (16 * 8), N4 / 64, 1);  // (128, 64)
    xgate_gemm<<<g1, blk, 0, stream>>>(x, W, bias, xg);

    for (int t = 0; t < T_DIM; ++t) {
        const float* hprev = (t == 0) ? h0 : out + (size_t)(t - 1) * B_DIM * H_DIM;
        lstm_step<<<dim3(H_DIM / 16), blk, 0, stream>>>(
            hprev, W,
            xg + (size_t)t * B_DIM * N4,
            reset + (size_t)t * B_DIM,
            c_state,
            out + (size_t)t * B_DIM * H_DIM);
    }

    float* hT = out + (size_t)T_DIM * B_DIM * H_DIM;
    float* cT = hT + (size_t)B_DIM * H_DIM;
    (void)hipMemcpyAsync(hT, out + (size_t)(T_DIM - 1) * B_DIM * H_DIM,
                         sizeof(float) * B_DIM * H_DIM, hipMemcpyDeviceToDevice, stream);
    (void)hipMemcpyAsync(cT, c_state,
                         sizeof(float) * B_DIM * H_DIM, hipMemcpyDeviceToDevice, stream);
}